// corr_block_83159156785843
// MI455X (gfx1250) — compile-verified
//
#include <hip/hip_runtime.h>

typedef __attribute__((ext_vector_type(2))) float v2f;
typedef __attribute__((ext_vector_type(8))) float v8f;

#define B_    4
#define C_    256
#define H_    96
#define W_    320
#define DMAX_ 64
#define KC_   32            // K-chunk staged in LDS per pipeline stage
#define LDSW_ 328           // padded row stride (floats): 320 + 8 breaks bank conflicts
#define NT_   640           // 20 waves of 32; wave id == row-tile index i (0..19)
#define BUFSZ_ (2 * KC_ * LDSW_)   // one stage: lf panel rows [0,KC), rf rows [KC,2KC)
#define NCHUNK_ (C_ / KC_)

// ---- async 16B global->LDS copy; %2 makes a pointer into LDS escape so the
// "memory" clobber forces LLVM to model the asm as writing LDS (else lds loads
// fold to undef). ----
__device__ __forceinline__ void async_copy16(unsigned ldsByte, const float* src,
                                             float* ldsPtr) {
  asm volatile("global_load_async_to_lds_b128 %0, %1, off"
               :: "v"(ldsByte), "v"(src), "v"(ldsPtr)
               : "memory");
}
__device__ __forceinline__ void wait_async0() {
  asm volatile("s_wait_asynccnt 0x0" ::: "memory");
}

// Stage one K-chunk (lf + rf panels, 2*KC_ rows x 320 floats) into LDS buffer.
__device__ __forceinline__ void stage_chunk(float* buf, unsigned bufByteBase,
                                            const float* lfBase, const float* rfBase,
                                            int c0, int tid) {
  const size_t cs = (size_t)H_ * W_;
#pragma unroll
  for (int q = 0; q < (2 * KC_ * 80) / NT_; ++q) {
    const int id  = q * NT_ + tid;
    const int row = id / 80;                 // 0..2*KC_-1
    const int col = (id - row * 80) * 4;     // float column, step 4
    const float* src = (row < KC_)
        ? (lfBase + (size_t)(c0 + row) * cs + col)
        : (rfBase + (size_t)(c0 + row - KC_) * cs + col);
    const unsigned off = (unsigned)(row * LDSW_ + col);
    async_copy16(bufByteBase + off * 4u, src, buf + off);
  }
}

// Compute one K-chunk: 8 K-steps of 4, up to 5 diagonal tiles per wave.
// FULL=true: branch-free (waves i>=4) so the scheduler can batch ds loads.
template <bool FULL>
__device__ __forceinline__ void compute_chunk(const float* buf, int iu, int m,
                                              int kl, v8f acc[5]) {
#pragma unroll
  for (int ks = 0; ks < KC_ / 4; ++ks) {
    const int k = 4 * ks + kl;
    v2f a;  // A(16x4): A[m][k] = lf[c0+k, 16*iu+m]
    a.x = buf[(k    ) * LDSW_ + 16 * iu + m];
    a.y = buf[(k + 1) * LDSW_ + 16 * iu + m];
#pragma unroll
    for (int dd = 0; dd < 5; ++dd) {
      if (!FULL && dd > iu) continue;       // scalar branch, partial waves only
      const int j = iu - dd;
      v2f bb;  // B(4x16): B[k][n] = rf[c0+k, 16*j+n], n == m
      bb.x = buf[(KC_ + k    ) * LDSW_ + 16 * j + m];
      bb.y = buf[(KC_ + k + 1) * LDSW_ + 16 * j + m];
      acc[dd] = __builtin_amdgcn_wmma_f32_16x16x4_f32(
          false, a, false, bb, (short)0, acc[dd], false, false);
    }
  }
}

// Banded-GEMM cost volume: out[b,d,h,w1] = (1/16) * sum_c lf[b,c,h,w1]*rf[b,c,h,w1-d]
// One workgroup per (b,h); wave i owns row tile i and diagonals D=0..4.
// Double-buffered async pipeline: chunk k+1 streams into LDS while chunk k computes.
__global__ __launch_bounds__(NT_) void corr_wmma_kernel(
    const float* __restrict__ lf, const float* __restrict__ rf,
    float* __restrict__ out) {
  __shared__ float lds[2 * BUFSZ_];        // two pipeline stages, ~164 KB total

  const int bh   = blockIdx.x;             // 0..383
  const int b    = bh / H_;
  const int h    = bh - b * H_;
  const int tid  = threadIdx.x;
  const int lane = tid & 31;
  const int iu   = __builtin_amdgcn_readfirstlane(tid >> 5);  // scalar wave id
  const int m    = lane & 15;
  const int kl   = (lane >> 4) << 1;       // lanes 0-15 -> K{0,1}, 16-31 -> K{2,3}

  const size_t cs = (size_t)H_ * W_;
  const float* lfBase = lf + (size_t)b * C_ * cs + (size_t)h * W_;
  const float* rfBase = rf + (size_t)b * C_ * cs + (size_t)h * W_;

  v8f acc[5];
#pragma unroll
  for (int dd = 0; dd < 5; ++dd) acc[dd] = v8f{0.f,0.f,0.f,0.f,0.f,0.f,0.f,0.f};

  // Prologue: start streaming chunk 0 into buffer 0.
  stage_chunk(lds, 0u, lfBase, rfBase, 0, tid);

  for (int k = 0; k < NCHUNK_; ++k) {
    wait_async0();        // this wave's chunk-k copies have landed in LDS
    __syncthreads();      // all waves' chunk-k landed; all done reading buf[(k+1)&1]
    if (k + 1 < NCHUNK_) {
      const int nb = (k + 1) & 1;
      stage_chunk(lds + nb * BUFSZ_, (unsigned)(nb * BUFSZ_) * 4u,
                  lfBase, rfBase, (k + 1) * KC_, tid);   // overlaps compute below
    }
    const float* buf = lds + (k & 1) * BUFSZ_;
    if (iu >= 4) compute_chunk<true >(buf, iu, m, kl, acc);
    else         compute_chunk<false>(buf, iu, m, kl, acc);
  }

  // ---- Scatter store: D-tile layout -> out[((b*64+d)*96+h)*320 + w1] ----
  // lane l, reg r holds (M = r + 8*(l>=16), N = l&15); d = 16*dd + M - N.
  // Tiles with j<0 were never accumulated -> acc==0 covers the padded region,
  // so every output element is written exactly once. Non-temporal: pure stream.
  const float scale = 0.0625f;  // 1/sqrt(256)
#pragma unroll
  for (int dd = 0; dd < 5; ++dd) {
#pragma unroll
    for (int r = 0; r < 8; ++r) {
      const int mm = r + ((lane >> 4) << 3);
      const int w1 = 16 * iu + mm;
      const int d  = 16 * dd + mm - m;
      if (d >= 0 && d < DMAX_) {
        __builtin_nontemporal_store(
            acc[dd][r] * scale,
            out + (((size_t)b * DMAX_ + d) * H_ + h) * W_ + w1);
      }
    }
  }
}

extern "C" void kernel_launch(void* const* d_in, const int* in_sizes, int n_in,
                              void* d_out, int out_size, void* d_ws, size_t ws_size,
                              hipStream_t stream) {
  const float* lf = (const float*)d_in[0];
  const float* rf = (const float*)d_in[1];
  // d_in[2] is d_max == 64, baked into the kernel.
  float* out = (float*)d_out;
  dim3 grid(B_ * H_);
  dim3 block(NT_);
  corr_wmma_kernel<<<grid, block, 0, stream>>>(lf, rf, out);
}